// GroupedVectorQuantizer_83133386981669
// MI455X (gfx1250) — compile-verified
//
#include <hip/hip_runtime.h>
#include <hip/hip_bf16.h>
#include <math.h>

// Problem constants (from reference)
#define BATCH   8192
#define GROUPS  8
#define KCODES  1024
#define DIM     64
// Output layout (floats): quantized(B*G*D) | indices(B*G) | 4 scalars
#define OFF_IDX   (BATCH * GROUPS * DIM)          // 4194304
#define OFF_SCAL  (OFF_IDX + BATCH * GROUPS)      // 4259840

typedef __attribute__((ext_vector_type(2))) float v2f;
typedef __attribute__((ext_vector_type(8))) float v8f;

// ---------------------------------------------------------------------------
// Kernel 0: zero the workspace accumulators (hist[1024] + sumsq[1]).
// Must run every launch: harness does not re-poison between graph replays.
// ---------------------------------------------------------------------------
__global__ void vq_init_kernel(float* __restrict__ hist, float* __restrict__ sumsq) {
    int t = blockIdx.x * blockDim.x + threadIdx.x;
    if (t < KCODES) hist[t] = 0.0f;
    if (t == 0) sumsq[0] = 0.0f;
}

// ---------------------------------------------------------------------------
// Kernel 1: per (16-row tile, group): fp32 WMMA distance matmul + argmin +
// gather + quantized write + commitment-loss partial + usage histogram.
// One workgroup = 8 waves; wave w owns codes [w*128, w*128+128).
// ---------------------------------------------------------------------------
__global__ __launch_bounds__(256) void vq_main_kernel(
    const float* __restrict__ z,        // (B, G*D)
    const float* __restrict__ codebook, // (G, K, D)
    float* __restrict__ out_q,          // (B, G*D)
    float* __restrict__ out_idx,        // (B, G) as float
    float* __restrict__ hist,           // (K,)
    float* __restrict__ sumsq)          // (1,)
{
    const int tile = blockIdx.x;           // 0..511  (16 batch rows each)
    const int g    = blockIdx.y;           // 0..7
    const int b0   = tile * 16;
    const int tid  = threadIdx.x;
    const int w    = tid >> 5;             // wave id 0..7
    const int lane = tid & 31;
    const int half = lane >> 4;            // 0: lanes 0-15, 1: lanes 16-31
    const int l16  = lane & 15;

    const float* cb = codebook + (size_t)g * KCODES * DIM;

    __shared__ __align__(16) float sA[16 * DIM];   // z tile (16 x 64), 4 KB
    __shared__ float candv[16][8];
    __shared__ int   candi[16][8];
    __shared__ int   sIdx[16];
    __shared__ float sSum;

    // ---- stage z tile into LDS (coalesced: 16 threads x float4 per row) ----
    {
        int row = tid >> 4;
        int c4  = (tid & 15) * 4;
        const float4 v = *(const float4*)(z + (size_t)(b0 + row) * (GROUPS * DIM)
                                            + g * DIM + c4);
        *(float4*)&sA[row * DIM + c4] = v;
    }
    if (tid == 0) sSum = 0.0f;
    __syncthreads();

    // ---- A fragments for V_WMMA_F32_16X16X4_F32, hoisted (reused 8x) ----
    // A is 16x4 per WMMA: lanes 0-15 hold row M=lane, K=(0,1); lanes 16-31
    // hold row M=lane-16, K=(2,3). 16 fragments cover Kdim=64.
    v2f afrag[16];
    {
        const float* arow = &sA[l16 * DIM + half * 2];
        #pragma unroll
        for (int j = 0; j < 16; ++j) {
            afrag[j].x = arow[4 * j + 0];
            afrag[j].y = arow[4 * j + 1];
        }
    }

    // Per-lane running argmin of (c2 - 2*zc) for rows r / r+8 (C-layout).
    float bestv[8];
    int   besti[8];
    #pragma unroll
    for (int r = 0; r < 8; ++r) { bestv[r] = 3.4e38f; besti[r] = 0; }

    for (int t = 0; t < 8; ++t) {
        const int n = w * 128 + t * 16 + l16;        // this lane's code column
        const float* cbn = cb + (size_t)n * DIM + half * 2;

        v8f  acc = {0.f, 0.f, 0.f, 0.f, 0.f, 0.f, 0.f, 0.f};
        float c2p = 0.0f;
        #pragma unroll
        for (int j = 0; j < 16; ++j) {
            // B is 4x16: lanes 0-15 hold col N=lane, K=(0,1); lanes 16-31 K=(2,3)
            v2f bfrag;
            bfrag.x = cbn[4 * j + 0];
            bfrag.y = cbn[4 * j + 1];
            c2p += bfrag.x * bfrag.x + bfrag.y * bfrag.y;
            acc = __builtin_amdgcn_wmma_f32_16x16x4_f32(
                      false, afrag[j], false, bfrag, (short)0, acc, false, false);
        }
        // full ||code||^2: combine the two half-wave K-partials
        const float c2 = c2p + __shfl_xor(c2p, 16, 32);

        // C/D layout: VGPR r = row r (lanes 0-15) / row r+8 (lanes 16-31)
        #pragma unroll
        for (int r = 0; r < 8; ++r) {
            const float dv = c2 - 2.0f * acc[r];
            const bool take = dv < bestv[r];          // earliest col wins ties
            bestv[r] = take ? dv : bestv[r];
            besti[r] = take ? n  : besti[r];
        }
    }

    // ---- argmin across the 16 lanes of each half-wave (shuffle tree) ----
    #pragma unroll
    for (int r = 0; r < 8; ++r) {
        float v = bestv[r];
        int   i = besti[r];
        #pragma unroll
        for (int o = 1; o < 16; o <<= 1) {
            const float ov = __shfl_xor(v, o, 32);
            const int   oi = __shfl_xor(i, o, 32);
            const bool take = (ov < v) || (ov == v && oi < i);
            v = take ? ov : v;
            i = take ? oi : i;
        }
        if (l16 == 0) {
            const int row = r + half * 8;
            candv[row][w] = v;
            candi[row][w] = i;
        }
    }
    __syncthreads();

    // ---- combine the 8 wave candidates per row (waves = ascending cols) ----
    if (tid < 16) {
        float v = candv[tid][0];
        int   i = candi[tid][0];
        #pragma unroll
        for (int wv = 1; wv < 8; ++wv) {
            const float ov = candv[tid][wv];
            const int   oi = candi[tid][wv];
            const bool take = (ov < v) || (ov == v && oi < i);
            v = take ? ov : v;
            i = take ? oi : i;
        }
        sIdx[tid] = i;
        out_idx[(size_t)(b0 + tid) * GROUPS + g] = (float)i;
        atomicAdd(&hist[i], 1.0f);       // exact in fp32 (counts <= 65536)
    }
    __syncthreads();

    // ---- gather winning code, write quantized, commitment-loss partial ----
    {
        const int row = tid >> 4;
        const int c4  = (tid & 15) * 4;
        const int idx = sIdx[row];
        const float4 qv = *(const float4*)(cb + (size_t)idx * DIM + c4);
        *(float4*)(out_q + (size_t)(b0 + row) * (GROUPS * DIM) + g * DIM + c4) = qv;
        const float d0 = sA[row * DIM + c4 + 0] - qv.x;
        const float d1 = sA[row * DIM + c4 + 1] - qv.y;
        const float d2 = sA[row * DIM + c4 + 2] - qv.z;
        const float d3 = sA[row * DIM + c4 + 3] - qv.w;
        atomicAdd(&sSum, d0 * d0 + d1 * d1 + d2 * d2 + d3 * d3);
    }
    __syncthreads();
    if (tid == 0) atomicAdd(sumsq, sSum);
}

// ---------------------------------------------------------------------------
// Kernel 2: entropy / perplexity / commitment loss scalars.
// ---------------------------------------------------------------------------
__global__ __launch_bounds__(256) void vq_finalize_kernel(
    const float* __restrict__ hist,
    const float* __restrict__ sumsq,
    float* __restrict__ out_scalars)
{
    __shared__ float red[256];
    const int tid = threadIdx.x;
    float e = 0.0f;
    for (int k = tid; k < KCODES; k += 256) {
        const float u = hist[k] * (1.0f / (float)(BATCH * GROUPS));
        e -= u * logf(u + 1e-10f);
    }
    red[tid] = e;
    __syncthreads();
    for (int s = 128; s > 0; s >>= 1) {
        if (tid < s) red[tid] += red[tid + s];
        __syncthreads();
    }
    if (tid == 0) {
        const float ent = red[0];
        out_scalars[0] = sumsq[0] * (1.0f / (float)(BATCH * GROUPS * DIM)); // commitment
        out_scalars[1] = 0.0f;                                              // codebook loss
        out_scalars[2] = ent;                                               // entropy
        out_scalars[3] = expf(ent);                                         // perplexity
    }
}

extern "C" void kernel_launch(void* const* d_in, const int* in_sizes, int n_in,
                              void* d_out, int out_size, void* d_ws, size_t ws_size,
                              hipStream_t stream) {
    const float* z        = (const float*)d_in[0];   // (B, G*D) fp32
    const float* codebook = (const float*)d_in[1];   // (G, K, D) fp32
    float* out = (float*)d_out;

    float* out_q    = out;             // quantized_flat
    float* out_idx  = out + OFF_IDX;   // indices (as float)
    float* out_scal = out + OFF_SCAL;  // commitment, codebook, entropy, perplexity

    float* hist  = (float*)d_ws;       // K floats
    float* sumsq = hist + KCODES;      // 1 float

    vq_init_kernel<<<dim3((KCODES + 255) / 256), dim3(256), 0, stream>>>(hist, sumsq);
    vq_main_kernel<<<dim3(BATCH / 16, GROUPS), dim3(256), 0, stream>>>(
        z, codebook, out_q, out_idx, hist, sumsq);
    vq_finalize_kernel<<<dim3(1), dim3(256), 0, stream>>>(hist, sumsq, out_scal);
}